// RotT_86071144612496
// MI455X (gfx1250) — compile-verified
//
#include <hip/hip_runtime.h>
#include <stdint.h>

#ifndef __has_builtin
#define __has_builtin(x) 0
#endif

// Problem geometry (compile-time constants from the reference module)
#define DRIGHT   2187      // 3^7 : element stride between j-planes (also d_left)
#define PLANE3   6561      // 3 * DRIGHT
#define NTRIP    4782969u  // 3^14 triples (a,b) pairs
#define BLOCK    256

// 8-byte vector for interleaved complex stores (clang ext vector so
// __builtin_nontemporal_store accepts it)
typedef __attribute__((ext_vector_type(2))) float f32x2;

// Builtin signature (from hipcc diagnostic): params are addrspace-qualified
// int* — global (AS1) source, LDS (AS3) destination.
typedef __attribute__((address_space(1))) int g1_int;
typedef __attribute__((address_space(3))) int l3_int;

#if __has_builtin(__builtin_amdgcn_global_load_async_to_lds_b32) && \
    __has_builtin(__builtin_amdgcn_s_wait_asynccnt)
#define USE_ASYNC_LDS 1
#else
#define USE_ASYNC_LDS 0
#endif

__device__ __forceinline__ l3_int* lds_cast(void* p) {
  // generic shared pointer: low 32 bits are the wave-relative LDS byte address
  return (l3_int*)(uint32_t)(uintptr_t)p;
}
__device__ __forceinline__ g1_int* glb_cast(const void* p) {
  return (g1_int*)(uintptr_t)p;
}

__global__ __launch_bounds__(BLOCK) void RotT_86071144612496_kernel(
    const float* __restrict__ xr, const float* __restrict__ xi,
    const float* __restrict__ ang, f32x2* __restrict__ out)
{
#if USE_ASYNC_LDS
  __shared__ float sm[6 * BLOCK];
#endif
  const unsigned tid = threadIdx.x;
  const unsigned t   = blockIdx.x * BLOCK + tid;
  if (t >= NTRIP) return;

  // t -> (a, b), base = a*6561 + b  (div by constant -> mulhi, cheap)
  const unsigned a    = t / DRIGHT;
  const unsigned b    = t - a * DRIGHT;
  const unsigned base = a * PLANE3 + b;

  // Uniform rotation coefficients: M = cos(th/2) I - i sin(th/2) sigma_x on {0,1}
  const float half = 0.5f * ang[0];
  float sn, cs;
  __sincosf(half, &sn, &cs);

  float x0r, x1r, x2r, x0i, x1i, x2i;

#if USE_ASYNC_LDS
  // CDNA5 async global->LDS copy path (ASYNCcnt). Each lane stages its own
  // 6 stream elements into lane-private LDS slots, waits, reads them back.
  __builtin_amdgcn_global_load_async_to_lds_b32(glb_cast(xr + base           ), lds_cast(&sm[0 * BLOCK + tid]), 0, 0);
  __builtin_amdgcn_global_load_async_to_lds_b32(glb_cast(xr + base +   DRIGHT), lds_cast(&sm[1 * BLOCK + tid]), 0, 0);
  __builtin_amdgcn_global_load_async_to_lds_b32(glb_cast(xr + base + 2*DRIGHT), lds_cast(&sm[2 * BLOCK + tid]), 0, 0);
  __builtin_amdgcn_global_load_async_to_lds_b32(glb_cast(xi + base           ), lds_cast(&sm[3 * BLOCK + tid]), 0, 0);
  __builtin_amdgcn_global_load_async_to_lds_b32(glb_cast(xi + base +   DRIGHT), lds_cast(&sm[4 * BLOCK + tid]), 0, 0);
  __builtin_amdgcn_global_load_async_to_lds_b32(glb_cast(xi + base + 2*DRIGHT), lds_cast(&sm[5 * BLOCK + tid]), 0, 0);
  __builtin_amdgcn_s_wait_asynccnt(0);
  x0r = sm[0 * BLOCK + tid];
  x1r = sm[1 * BLOCK + tid];
  x2r = sm[2 * BLOCK + tid];
  x0i = sm[3 * BLOCK + tid];
  x1i = sm[4 * BLOCK + tid];
  x2i = sm[5 * BLOCK + tid];
#else
  // Fallback: NT direct loads + gfx1250 prefetch (global_prefetch_b8)
  __builtin_prefetch(xr + base, 0, 0);
  __builtin_prefetch(xi + base, 0, 0);
  x0r = __builtin_nontemporal_load(xr + base);
  x1r = __builtin_nontemporal_load(xr + base +   DRIGHT);
  x2r = __builtin_nontemporal_load(xr + base + 2*DRIGHT);
  x0i = __builtin_nontemporal_load(xi + base);
  x1i = __builtin_nontemporal_load(xi + base +   DRIGHT);
  x2i = __builtin_nontemporal_load(xi + base + 2*DRIGHT);
#endif

  // y0 = c*x0 - i s x1 ; y1 = -i s x0 + c x1 ; y2 = x2
  f32x2 y0, y1, y2;
  y0.x = cs * x0r + sn * x1i;
  y0.y = cs * x0i - sn * x1r;
  y1.x = sn * x0i + cs * x1r;
  y1.y = cs * x1i - sn * x0r;
  y2.x = x2r;
  y2.y = x2i;

  // Interleaved complex64 output, 8B-aligned NT stores (once-written stream)
  __builtin_nontemporal_store(y0, out + base);
  __builtin_nontemporal_store(y1, out + base +   DRIGHT);
  __builtin_nontemporal_store(y2, out + base + 2*DRIGHT);
}

extern "C" void kernel_launch(void* const* d_in, const int* in_sizes, int n_in,
                              void* d_out, int out_size, void* d_ws, size_t ws_size,
                              hipStream_t stream) {
  (void)in_sizes; (void)n_in; (void)out_size; (void)d_ws; (void)ws_size;
  const float* xr  = (const float*)d_in[0];   // x_real, D floats
  const float* xi  = (const float*)d_in[1];   // x_imag, D floats
  const float* ang = (const float*)d_in[2];   // angle, 1 float
  f32x2* out = (f32x2*)d_out;                 // complex64 -> interleaved float pairs

  const unsigned blocks = (NTRIP + BLOCK - 1) / BLOCK;  // 18684
  RotT_86071144612496_kernel<<<blocks, BLOCK, 0, stream>>>(xr, xi, ang, out);
}